// EuclideanCodebook_44856638440088
// MI455X (gfx1250) — compile-verified
//
#include <hip/hip_runtime.h>

// ---------------------------------------------------------------------------
// EuclideanCodebook for MI455X (gfx1250, wave32, WMMA, async global->LDS).
//   dist = xsq + esq - 2*x@e^T   (fp32 via V_WMMA_F32_16X16X4_F32)
//   argmin per row: packed (distbits,col) u64, LDS ds_min_u64 combine,
//                   one global atomicMin per row per block
//   quantize = gather(embed, argmin)
// Async staging uses GLOBAL_LOAD_ASYNC_TO_LDS_B128 via inline asm (the clang
// builtin's params are LangAS-qualified and unspellable from HIP source).
// ---------------------------------------------------------------------------

typedef __attribute__((ext_vector_type(2))) float v2f;
typedef __attribute__((ext_vector_type(8))) float v8f;

#define N_TOK 32768
#define DIM   256
#define KCB   4096

#define M_BLK 64          // token rows per block
#define C_BLK 128         // codebook cols per block
#define DC    64          // D-chunk staged in LDS per iteration
#define PITCH 68          // LDS row pitch in floats (64+4 pad: conflict-free frags)
#define CHUNKS (DIM / DC) // 4
// per-thread async b128 loads per chunk: X 4 + E 8 = 12

// Generic shared pointer -> 32-bit LDS byte offset (LDS aperture occupies the
// high 32 bits of the flat address; low 32 bits are the wave-relative offset).
__device__ __forceinline__ unsigned lds_off_u32(const void* p) {
  return (unsigned)(unsigned long long)p;
}

// 16-byte global -> LDS direct DMA, tracked by ASYNCcnt. VDST = LDS offset,
// VADDR = 64-bit global address, no SADDR.
__device__ __forceinline__ void cp16_async(const float* __restrict__ g, float* l) {
  asm volatile("global_load_async_to_lds_b128 %0, %1, off"
               :: "v"(lds_off_u32(l)), "v"(g) : "memory");
}
// Async loads complete in issue order: waiting for <=12 retires the older of
// two 12-load chunks while the newer one is still in flight.
#define ASYNC_WAIT_OLDER() asm volatile("s_wait_asynccnt 12" ::: "memory")
#define ASYNC_WAIT_ALL()   asm volatile("s_wait_asynccnt 0"  ::: "memory")

// ---------------------------------------------------------------------------
__global__ void init_keys_kernel(unsigned long long* __restrict__ keys, int n) {
  int i = blockIdx.x * blockDim.x + threadIdx.x;
  if (i < n) keys[i] = 0xFFFFFFFFFFFFFFFFull;
}

// One wave per row: 32 lanes x 8 floats = 256 = DIM. 8 rows per 256-thread block.
__global__ void row_sqnorm_kernel(const float* __restrict__ src,
                                  float* __restrict__ out, int rows) {
  int wave = threadIdx.x >> 5;
  int lane = threadIdx.x & 31;
  int row  = blockIdx.x * 8 + wave;
  if (row >= rows) return;
  const float4* p = (const float4*)(src + (size_t)row * DIM) + lane * 2;
  float4 a = p[0], b = p[1];
  float s = a.x*a.x + a.y*a.y + a.z*a.z + a.w*a.w
          + b.x*b.x + b.y*b.y + b.z*b.z + b.w*b.w;
  #pragma unroll
  for (int off = 16; off; off >>= 1) s += __shfl_xor(s, off);
  if (lane == 0) out[row] = s;
}

// ---------------------------------------------------------------------------
// Main GEMM + epilogue. Block = 256 threads = 8 wave32.
// Wave w owns cols [c0+16w, c0+16w+16) for all 64 rows (4 M-tiles).
// Double-buffered LDS; async DMA keeps one chunk in flight under WMMA.
__global__ void __launch_bounds__(256, 2)
dist_gemm_kernel(const float* __restrict__ x, const float* __restrict__ e,
                 const float* __restrict__ xsq, const float* __restrict__ esq,
                 float* __restrict__ dist, unsigned long long* __restrict__ keys) {
  __shared__ float Xs[2][M_BLK * PITCH];            // 2 x 17.4 KB
  __shared__ float Es[2][C_BLK * PITCH];            // 2 x 34.8 KB
  __shared__ unsigned long long rkey[M_BLK];        // per-row argmin keys

  const int t    = threadIdx.x;
  const int wave = t >> 5;
  const int lane = t & 31;
  const int hl   = lane >> 4;
  const int lr   = lane & 15;
  const int m0   = blockIdx.x * M_BLK;
  const int c0   = blockIdx.y * C_BLK;

  if (t < M_BLK) rkey[t] = 0xFFFFFFFFFFFFFFFFull;   // covered by loop barriers

  v8f acc[4];
  #pragma unroll
  for (int i = 0; i < 4; ++i)
    #pragma unroll
    for (int j = 0; j < 8; ++j) acc[i][j] = 0.0f;

  const int srow = t >> 4;        // 0..15
  const int sc   = (t & 15) * 4;  // float column within 64-wide chunk

  auto stage = [&](int c, int b) {
    const int d0 = c * DC;
    #pragma unroll
    for (int i = 0; i < 4; ++i) {   // X: 64 rows, 4 b128/thread
      int r = srow + i * 16;
      cp16_async(x + (size_t)(m0 + r) * DIM + d0 + sc, &Xs[b][r * PITCH + sc]);
    }
    #pragma unroll
    for (int i = 0; i < 8; ++i) {   // E: 128 rows, 8 b128/thread
      int r = srow + i * 16;
      cp16_async(e + (size_t)(c0 + r) * DIM + d0 + sc, &Es[b][r * PITCH + sc]);
    }
  };

  stage(0, 0);
  stage(1, 1);

  #pragma unroll
  for (int c = 0; c < CHUNKS; ++c) {
    const int b = c & 1;
    if (c + 1 < CHUNKS) ASYNC_WAIT_OLDER();   // chunk c landed, c+1 may fly
    else                ASYNC_WAIT_ALL();
    __syncthreads();                          // all waves' chunk c in LDS

    // V_WMMA_F32_16X16X4_F32 frags (ISA 7.12.2):
    //   A 16x4 / B 4x16: lane(l%16)=M/N, VGPR pair = K {2*(l/16), 2*(l/16)+1}
    #pragma unroll
    for (int kk = 0; kk < DC; kk += 4) {
      v2f bf = *(const v2f*)(&Es[b][(wave * 16 + lr) * PITCH + kk + 2 * hl]);
      #pragma unroll
      for (int mt = 0; mt < 4; ++mt) {
        v2f af = *(const v2f*)(&Xs[b][(mt * 16 + lr) * PITCH + kk + 2 * hl]);
        acc[mt] = __builtin_amdgcn_wmma_f32_16x16x4_f32(
            false, af, false, bf, (short)0, acc[mt], false, false);
      }
    }
    __syncthreads();                          // everyone done reading buffer b
    if (c + 2 < CHUNKS) stage(c + 2, b);      // refill it via async DMA
  }

  // Epilogue. C/D layout: VGPR v, lanes 0-15 -> M=v, lanes 16-31 -> M=v+8, N=l%16.
  const int col = c0 + wave * 16 + lr;
  const float es_c = esq[col];
  #pragma unroll
  for (int mt = 0; mt < 4; ++mt) {
    #pragma unroll
    for (int v = 0; v < 8; ++v) {
      const int rl = mt * 16 + hl * 8 + v;                 // local row 0..63
      float dval = fmaxf(fmaf(-2.0f, acc[mt][v], xsq[m0 + rl] + es_c), 0.0f);
      // 536 MB write-once stream: keep it out of L2
      __builtin_nontemporal_store(dval, dist + (size_t)(m0 + rl) * KCB + col);

      // argmin key: dist >= 0 so float bits are order-preserving; ties pick
      // the smallest col (matches argmin first-occurrence).
      unsigned long long key =
          ((unsigned long long)__float_as_uint(dval) << 32) | (unsigned int)col;
      #pragma unroll
      for (int off = 1; off < 16; off <<= 1) {
        unsigned long long o = __shfl_xor(key, off);
        key = (o < key) ? o : key;
      }
      if (lr == 0) atomicMin(&rkey[rl], key);              // ds_min_u64 on WGP
    }
  }
  __syncthreads();
  if (t < M_BLK) atomicMin(keys + m0 + t, rkey[t]);        // 1 global atomic/row
}

// ---------------------------------------------------------------------------
// 4 rows per 256-thread block; 64 threads gather one 256-float codeword.
__global__ void finalize_kernel(const unsigned long long* __restrict__ keys,
                                const float* __restrict__ e,
                                float* __restrict__ quant,
                                float* __restrict__ ind_out) {
  const int row = blockIdx.x * 4 + (threadIdx.x >> 6);
  const int tq  = threadIdx.x & 63;
  const unsigned long long key = keys[row];
  const unsigned int idx = (unsigned int)(key & 0xFFFFFFFFull);
  if (tq == 0) ind_out[row] = (float)idx;
  float4 v = *(const float4*)(e + (size_t)idx * DIM + tq * 4);
  *(float4*)(quant + (size_t)row * DIM + tq * 4) = v;
}

// ---------------------------------------------------------------------------
extern "C" void kernel_launch(void* const* d_in, const int* in_sizes, int n_in,
                              void* d_out, int out_size, void* d_ws, size_t ws_size,
                              hipStream_t stream) {
  const float* x = (const float*)d_in[0];   // [N, D]
  const float* e = (const float*)d_in[1];   // [1, K, D] -> flat [K, D]

  // d_out: quantize [N,D] | embed_ind [N] (as float) | dist [N,K]
  float* quant   = (float*)d_out;
  float* ind_out = quant + (size_t)N_TOK * DIM;
  float* dist    = ind_out + N_TOK;

  // workspace: keys u64[N] | xsq f32[N] | esq f32[K]  (~410 KB)
  unsigned long long* keys = (unsigned long long*)d_ws;
  float* xsq = (float*)((char*)d_ws + (size_t)N_TOK * 8);
  float* esq = xsq + N_TOK;

  init_keys_kernel<<<N_TOK / 256, 256, 0, stream>>>(keys, N_TOK);
  row_sqnorm_kernel<<<N_TOK / 8, 256, 0, stream>>>(x, xsq, N_TOK);
  row_sqnorm_kernel<<<KCB / 8, 256, 0, stream>>>(e, esq, KCB);

  dim3 grid(N_TOK / M_BLK, KCB / C_BLK);   // 512 x 32
  dist_gemm_kernel<<<grid, 256, 0, stream>>>(x, e, xsq, esq, dist, keys);

  finalize_kernel<<<N_TOK / 4, 256, 0, stream>>>(keys, e, quant, ind_out);
}